// PosController_Vectorized_19146964205853
// MI455X (gfx1250) — compile-verified
//
#include <hip/hip_runtime.h>
#include <math.h>

// Elementwise PID position controller. Memory-bound stream:
//   in  = 68 B/elem, out = 12 B/elem, N = 4.19M  -> ~335 MB @ 23.3 TB/s ~ 14.4 us floor.
// Strategy: 4 elements per thread so every access is a coalesced 16B b128
// load/store with non-temporal hints (one-touch stream, no L2 pollution).
// 32-bit per-lane offsets -> saddr+voffset global loads (less VALU, fewer VGPRs).

typedef float v4f __attribute__((ext_vector_type(4)));

#define G_ACC      9.81f
#define DT_VEL     0.01f
#define INV_DT     100.0f
#define MAX_THRUST (0.9f * 44.4f)
#define MAX_ANGLE  (30.0f * 3.14159265358979323846f / 180.0f)
#define TAU_       0.05f

__device__ __forceinline__ void pid_one(
    const float err_in[3], const float integ_in[3],
    const float pe[3], const float pd[3],
    float yaw, float m,
    float& rp0_out, float& rp1_out, float& thr_out)
{
    const float kp[3] = {2.0f, 2.0f, 4.0f};
    const float ki[3] = {0.5f, 0.5f, 1.0f};
    const float kd[3] = {0.1f, 0.1f, 0.05f};
    const float alpha = DT_VEL / (TAU_ + DT_VEL);

    float u[3];
#pragma unroll
    for (int c = 0; c < 3; ++c) {
        float err   = err_in[c];
        float integ = integ_in[c] + err * DT_VEL;
        float draw  = (err - pe[c]) * INV_DT;
        float d     = pd[c] + alpha * (draw - pd[c]);
        float uu    = kp[c] * err + ki[c] * integ + kd[c] * d;
        u[c] = fminf(fmaxf(uu, -6.0f), 6.0f);
    }

    float s, c;
    __sincosf(yaw, &s, &c);
    float rp0 = (s * u[0] - c * u[1]) / G_ACC;
    float rp1 = (c * u[0] + s * u[1]) / G_ACC;
    float mag = sqrtf(rp0 * rp0 + rp1 * rp1);
    float scale = (mag > MAX_ANGLE) ? fminf(MAX_ANGLE / (mag + 1e-6f), 1.0f)
                                    : 1.0f;
    rp0_out = rp0 * scale;
    rp1_out = rp1 * scale;

    float thr = m * G_ACC + m * u[2];
    thr_out = fminf(fmaxf(thr, 0.8f * G_ACC * m), MAX_THRUST);
}

__global__ __launch_bounds__(256)
void pid_ctrl_kernel(const float* __restrict__ ref_ve,
                     const float* __restrict__ meas_ve,
                     const float* __restrict__ meas_yaw,
                     const float* __restrict__ mass,
                     const float* __restrict__ pid_int,
                     const float* __restrict__ pid_prev_err,
                     const float* __restrict__ pid_prev_d,
                     float* __restrict__ out_rp,
                     float* __restrict__ out_thr,
                     unsigned n4)
{
    const unsigned t = blockIdx.x * blockDim.x + threadIdx.x;
    if (t >= n4) return;

    // 32-bit element offsets: max byte offset 12*n4*4 ~ 50 MB, fits easily.
    const unsigned t3 = 3u * t;

    // ---- vector loads: 12 consecutive floats (4 elems x 3 comps) per (N,3) array
    v4f rv[3], mv[3], iv[3], pev[3], pdv[3];
    {
        const v4f* p;
        p = (const v4f*)ref_ve + t3;
        rv[0] = __builtin_nontemporal_load(p + 0);
        rv[1] = __builtin_nontemporal_load(p + 1);
        rv[2] = __builtin_nontemporal_load(p + 2);
        p = (const v4f*)meas_ve + t3;
        mv[0] = __builtin_nontemporal_load(p + 0);
        mv[1] = __builtin_nontemporal_load(p + 1);
        mv[2] = __builtin_nontemporal_load(p + 2);
        p = (const v4f*)pid_int + t3;
        iv[0] = __builtin_nontemporal_load(p + 0);
        iv[1] = __builtin_nontemporal_load(p + 1);
        iv[2] = __builtin_nontemporal_load(p + 2);
        p = (const v4f*)pid_prev_err + t3;
        pev[0] = __builtin_nontemporal_load(p + 0);
        pev[1] = __builtin_nontemporal_load(p + 1);
        pev[2] = __builtin_nontemporal_load(p + 2);
        p = (const v4f*)pid_prev_d + t3;
        pdv[0] = __builtin_nontemporal_load(p + 0);
        pdv[1] = __builtin_nontemporal_load(p + 1);
        pdv[2] = __builtin_nontemporal_load(p + 2);
    }
    const v4f yv = __builtin_nontemporal_load((const v4f*)meas_yaw + t);
    const v4f mm = __builtin_nontemporal_load((const v4f*)mass + t);

    // ---- unpack to per-element scalars (fully unrolled -> stays in VGPRs)
    float fe[12], fi[12], fpe[12], fpd[12];
#pragma unroll
    for (int k = 0; k < 12; ++k) {
        fe[k]  = rv[k >> 2][k & 3] - mv[k >> 2][k & 3]; // err = ref - meas
        fi[k]  = iv[k >> 2][k & 3];
        fpe[k] = pev[k >> 2][k & 3];
        fpd[k] = pdv[k >> 2][k & 3];
    }

    v4f rp_lo, rp_hi, thr;
#pragma unroll
    for (int e = 0; e < 4; ++e) {
        const float err3[3] = { fe[3*e],  fe[3*e+1],  fe[3*e+2]  };
        const float int3[3] = { fi[3*e],  fi[3*e+1],  fi[3*e+2]  };
        const float pe3[3]  = { fpe[3*e], fpe[3*e+1], fpe[3*e+2] };
        const float pd3[3]  = { fpd[3*e], fpd[3*e+1], fpd[3*e+2] };
        float rp0, rp1, th;
        pid_one(err3, int3, pe3, pd3, yv[e], mm[e], rp0, rp1, th);
        if (e < 2) { rp_lo[2*e] = rp0;       rp_lo[2*e+1] = rp1;       }
        else       { rp_hi[2*(e-2)] = rp0;   rp_hi[2*(e-2)+1] = rp1;   }
        thr[e] = th;
    }

    // ---- vector stores: rp pairs are 8 consecutive floats, thrust 4
    v4f* orp = (v4f*)out_rp + 2u * t;
    __builtin_nontemporal_store(rp_lo, orp + 0);
    __builtin_nontemporal_store(rp_hi, orp + 1);
    __builtin_nontemporal_store(thr, (v4f*)out_thr + t);
}

// Scalar tail for N % 4 != 0 (never launched for the harness N = 4194304;
// launch decision is a pure function of in_sizes, so graph-capture safe).
__global__ void pid_ctrl_tail_kernel(const float* __restrict__ ref_ve,
                                     const float* __restrict__ meas_ve,
                                     const float* __restrict__ meas_yaw,
                                     const float* __restrict__ mass,
                                     const float* __restrict__ pid_int,
                                     const float* __restrict__ pid_prev_err,
                                     const float* __restrict__ pid_prev_d,
                                     float* __restrict__ out_rp,
                                     float* __restrict__ out_thr,
                                     unsigned base, unsigned n)
{
    const unsigned i = base + blockIdx.x * blockDim.x + threadIdx.x;
    if (i >= n) return;
    float err3[3], int3[3], pe3[3], pd3[3];
    for (int c = 0; c < 3; ++c) {
        err3[c] = ref_ve[3*i + c] - meas_ve[3*i + c];
        int3[c] = pid_int[3*i + c];
        pe3[c]  = pid_prev_err[3*i + c];
        pd3[c]  = pid_prev_d[3*i + c];
    }
    float rp0, rp1, th;
    pid_one(err3, int3, pe3, pd3, meas_yaw[i], mass[i], rp0, rp1, th);
    out_rp[2*i]     = rp0;
    out_rp[2*i + 1] = rp1;
    out_thr[i]      = th;
}

extern "C" void kernel_launch(void* const* d_in, const int* in_sizes, int n_in,
                              void* d_out, int out_size, void* d_ws, size_t ws_size,
                              hipStream_t stream)
{
    const float* ref_ve       = (const float*)d_in[0];
    const float* meas_ve      = (const float*)d_in[1];
    const float* meas_yaw     = (const float*)d_in[2];
    const float* mass         = (const float*)d_in[3];
    const float* pid_int      = (const float*)d_in[4];
    const float* pid_prev_err = (const float*)d_in[5];
    const float* pid_prev_d   = (const float*)d_in[6];
    float* out     = (float*)d_out;
    const unsigned N   = (unsigned)in_sizes[2];   // meas_yaw is (N,)
    float* out_thr = out + 2ull * N;              // tuple layout: [rp (N,2)] [thrust (N,)]

    const unsigned n4    = N >> 2;
    const int      block = 256;                   // 8 wave32 per block
    if (n4 > 0) {
        const unsigned grid = (n4 + block - 1) / block;
        pid_ctrl_kernel<<<grid, block, 0, stream>>>(
            ref_ve, meas_ve, meas_yaw, mass, pid_int, pid_prev_err, pid_prev_d,
            out, out_thr, n4);
    }
    const unsigned tail = N & 3u;
    if (tail) {
        pid_ctrl_tail_kernel<<<1, 32, 0, stream>>>(
            ref_ve, meas_ve, meas_yaw, mass, pid_int, pid_prev_err, pid_prev_d,
            out, out_thr, N - tail, N);
    }
}